// Chamfer_Loss_8993661518576
// MI455X (gfx1250) — compile-verified
//
#include <hip/hip_runtime.h>
#include <hip/hip_bf16.h>

// ---------------------------------------------------------------------------
// Mesh loss for MI455X (gfx1250, wave32).
// Chamfer dominates: d(i,j) = xx_i + yy_j - 2 <x_i, y_j>; the 3-D dot
// products are done 16x16 per wave with V_WMMA_F32_16X16X4_F32 (K pad 3->4),
// full f32 (chamfer takes a min -> no low-precision shortcut).
// Points are pre-packed as float4 {x,y,z,|p|^2}, padded with {0,0,0,BIG} to a
// multiple of 16, so the WMMA inner loop is branch-free: one unconditional
// b128 load per lane per 16-column tile (the packed region carries one tile
// of slack so the software-pipelined next-tile load needs no guard).
// ---------------------------------------------------------------------------

typedef float v2f __attribute__((ext_vector_type(2)));
typedef float v8f __attribute__((ext_vector_type(8)));

#define W_EDGE   0.5f
#define W_LAP    0.05f
#define W_NORMAL 0.01f
#define W_VEL    10.0f
#define BATCH    2
#define BIGF     3.0e38f

// ---- helpers --------------------------------------------------------------

__device__ __forceinline__ void wave_atomic_add(float* dst, float v) {
#pragma unroll
  for (int off = 16; off >= 1; off >>= 1)
    v += __shfl_xor(v, off, 32);
  if ((threadIdx.x & 31) == 0) atomicAdd(dst, v);
}

// ---- zero workspace -------------------------------------------------------

__global__ void zero_kernel(float* __restrict__ ws, int n) {
  int i = blockIdx.x * blockDim.x + threadIdx.x;
  if (i < n) ws[i] = 0.0f;
}

// ---- pack points: {x,y,z,|p|^2}, padded with {0,0,0,BIG} ------------------
// diff!=0: velocity mode, point i = p[i+1]-p[i] (n = Norig-1).
// npad includes one extra tile (16 entries) of slack for the pipelined load.

__global__ void pack_points(const float* __restrict__ src,
                            float4* __restrict__ dst,
                            int n, int npad, int strideOrig, int diff) {
  int i = blockIdx.x * blockDim.x + threadIdx.x;
  int b = blockIdx.y;
  if (i >= npad) return;
  const float* p = src + (size_t)b * strideOrig * 3;
  float4 o;
  if (i < n) {
    float c0, c1, c2;
    if (diff) {
      c0 = p[(i + 1) * 3 + 0] - p[i * 3 + 0];
      c1 = p[(i + 1) * 3 + 1] - p[i * 3 + 1];
      c2 = p[(i + 1) * 3 + 2] - p[i * 3 + 2];
    } else {
      c0 = p[i * 3 + 0];
      c1 = p[i * 3 + 1];
      c2 = p[i * 3 + 2];
    }
    o.x = c0; o.y = c1; o.z = c2;
    o.w = c0 * c0 + c1 * c1 + c2 * c2;
  } else {
    o.x = 0.0f; o.y = 0.0f; o.z = 0.0f; o.w = BIGF;  // excluded from min
  }
  dst[(size_t)b * npad + i] = o;
}

// ---- chamfer row-min via WMMA f32 16x16x4 ---------------------------------
// One wave owns a 16-row strip of X, scans all column tiles of Y, and
// accumulates sum_i(min_j d) * scale. grid.y = batch.
// ntiles excludes the slack tile; the pipelined load may touch it (value dead).

__global__ void chamfer_rowmin_wmma(const float4* __restrict__ Xp,
                                    const float4* __restrict__ Yp,
                                    int n, int npad,
                                    float* __restrict__ accum, float scale) {
  const int lane  = threadIdx.x & 31;
  const int wave  = threadIdx.x >> 5;
  const int wpb   = blockDim.x >> 5;
  const int rowBase = (blockIdx.x * wpb + wave) * 16;
  if (rowBase >= n) return;  // uniform per wave -> EXEC all-ones at WMMA

  const int b = blockIdx.y;
  const float4* x = Xp + (size_t)b * npad;
  const float4* y = Yp + (size_t)b * npad;

  const int half = lane >> 4;   // 0: K 0..1 | 1: K 2..3 (pad)
  const int l15  = lane & 15;

  // A matrix 16x4 f32: row M = l15, VGPR v holds K = 2*half + v (K=3 -> 0)
  const float4 ax = x[rowBase + l15];
  v2f A;
  A.x = half ? ax.z : ax.x;
  A.y = half ? 0.0f : ax.y;

  // xx for the 8 D-rows this lane sees (M = r + 8*half): shuffle from the
  // float4s already held by lanes 0..15.
  float xx[8];
#pragma unroll
  for (int r = 0; r < 8; ++r)
    xx[r] = __shfl(ax.w, r + 8 * half, 32);

  float minv[8];
#pragma unroll
  for (int r = 0; r < 8; ++r) minv[r] = BIGF;

  const int ntiles = (npad >> 4) - 1;  // last tile is pipeline slack
  float4 cur = y[l15];
  for (int t = 0; t < ntiles; ++t) {
    // Unconditional pipelined load; t==ntiles-1 reads the slack tile (dead).
    float4 nxt = y[(t + 1) * 16 + l15];

    // B matrix 4x16 f32: N = l15, VGPR v holds K = 2*half + v
    v2f Bm;
    Bm.x = half ? cur.z : cur.x;
    Bm.y = half ? 0.0f : cur.y;
    const float yy = cur.w;

    v8f D = {};
    D = __builtin_amdgcn_wmma_f32_16x16x4_f32(
        /*neg_a=*/false, A, /*neg_b=*/false, Bm,
        /*c_mod=*/(short)0, D, /*reuse_a=*/false, /*reuse_b=*/false);

#pragma unroll
    for (int r = 0; r < 8; ++r) {
      float d = fmaxf(xx[r] + yy - 2.0f * D[r], 0.0f);
      minv[r] = fminf(minv[r], d);
    }
    cur = nxt;
  }

  // min across the 16 lanes of each half (xor offsets < 16 keep halves apart)
#pragma unroll
  for (int r = 0; r < 8; ++r) {
    float v = minv[r];
#pragma unroll
    for (int off = 8; off >= 1; off >>= 1)
      v = fminf(v, __shfl_xor(v, off, 32));
    minv[r] = v;
  }
  if (l15 == 0) {
#pragma unroll
    for (int r = 0; r < 8; ++r) {
      int row = rowBase + r + 8 * half;
      if (row < n) atomicAdd(accum, minv[r] * scale);
    }
  }
}

// ---- edge loss ------------------------------------------------------------

__global__ void edge_loss_kernel(const float* __restrict__ pred,
                                 const int* __restrict__ edges,
                                 int E, int N, float* __restrict__ accum,
                                 float scale) {
  int e = blockIdx.x * blockDim.x + threadIdx.x;
  int b = blockIdx.y;
  float val = 0.0f;
  if (e < E) {
    const float* v = pred + (size_t)b * N * 3;
    int a = edges[e * 2 + 0], c = edges[e * 2 + 1];
    float dx = v[a * 3 + 0] - v[c * 3 + 0];
    float dy = v[a * 3 + 1] - v[c * 3 + 1];
    float dz = v[a * 3 + 2] - v[c * 3 + 2];
    val = (dx * dx + dy * dy + dz * dz) * scale;
  }
  wave_atomic_add(accum, val);
}

// ---- cotangent Laplacian: scatter phase -----------------------------------

__global__ void lap_scatter(const float* __restrict__ pred,
                            const int* __restrict__ faces,
                            int F, int N, float* __restrict__ Lx,
                            float* __restrict__ rowsum) {
  int f = blockIdx.x * blockDim.x + threadIdx.x;
  int b = blockIdx.y;
  if (f >= F) return;
  const float* v = pred + (size_t)b * N * 3;
  float* lx = Lx + (size_t)b * N * 3;
  float* rs = rowsum + (size_t)b * N;

  int i0 = faces[f * 3 + 0], i1 = faces[f * 3 + 1], i2 = faces[f * 3 + 2];
  float p0x = v[i0 * 3], p0y = v[i0 * 3 + 1], p0z = v[i0 * 3 + 2];
  float p1x = v[i1 * 3], p1y = v[i1 * 3 + 1], p1z = v[i1 * 3 + 2];
  float p2x = v[i2 * 3], p2y = v[i2 * 3 + 1], p2z = v[i2 * 3 + 2];

  float A2 = (p1x - p2x) * (p1x - p2x) + (p1y - p2y) * (p1y - p2y) + (p1z - p2z) * (p1z - p2z);
  float B2 = (p0x - p2x) * (p0x - p2x) + (p0y - p2y) * (p0y - p2y) + (p0z - p2z) * (p0z - p2z);
  float C2 = (p0x - p1x) * (p0x - p1x) + (p0y - p1y) * (p0y - p1y) + (p0z - p1z) * (p0z - p1z);
  float Al = sqrtf(A2), Bl = sqrtf(B2), Cl = sqrtf(C2);
  float s = 0.5f * (Al + Bl + Cl);
  float area = sqrtf(fmaxf(s * (s - Al) * (s - Bl) * (s - Cl), 1e-12f));
  float inv4a = 0.25f / area;
  float cota = (B2 + C2 - A2) * inv4a;
  float cotb = (A2 + C2 - B2) * inv4a;
  float cotc = (A2 + B2 - C2) * inv4a;

  // (i,j,w) entries from ii=[1,2,0], jj=[2,0,1], plus symmetric swap.
  int ii[3] = {i1, i2, i0};
  int jj[3] = {i2, i0, i1};
  float ww[3] = {cota, cotb, cotc};
#pragma unroll
  for (int t = 0; t < 3; ++t) {
    int i = ii[t], j = jj[t];
    float w = ww[t];
    atomicAdd(&lx[i * 3 + 0], w * v[j * 3 + 0]);
    atomicAdd(&lx[i * 3 + 1], w * v[j * 3 + 1]);
    atomicAdd(&lx[i * 3 + 2], w * v[j * 3 + 2]);
    atomicAdd(&rs[i], w);
    atomicAdd(&lx[j * 3 + 0], w * v[i * 3 + 0]);
    atomicAdd(&lx[j * 3 + 1], w * v[i * 3 + 1]);
    atomicAdd(&lx[j * 3 + 2], w * v[i * 3 + 2]);
    atomicAdd(&rs[j], w);
  }
}

// ---- cotangent Laplacian: gather/finish phase -----------------------------

__global__ void lap_finish(const float* __restrict__ pred,
                           const float* __restrict__ Lx,
                           const float* __restrict__ rowsum,
                           int N, float* __restrict__ accum, float scale) {
  int i = blockIdx.x * blockDim.x + threadIdx.x;
  int b = blockIdx.y;
  float val = 0.0f;
  if (i < N) {
    const float* v = pred + (size_t)b * N * 3;
    const float* lx = Lx + (size_t)b * N * 3;
    float rs = rowsum[(size_t)b * N + i];
    float nw = rs > 0.0f ? 1.0f / rs : 0.0f;
    float rx = lx[i * 3 + 0] * nw - v[i * 3 + 0];
    float ry = lx[i * 3 + 1] * nw - v[i * 3 + 1];
    float rz = lx[i * 3 + 2] * nw - v[i * 3 + 2];
    val = sqrtf(rx * rx + ry * ry + rz * rz) * scale;
  }
  wave_atomic_add(accum, val);
}

// ---- normal consistency ---------------------------------------------------

__global__ void nc_kernel(const float* __restrict__ pred,
                          const int* __restrict__ pairs,
                          int P, int N, float* __restrict__ accum,
                          float scale) {
  int p = blockIdx.x * blockDim.x + threadIdx.x;
  int b = blockIdx.y;
  float val = 0.0f;
  if (p < P) {
    const float* v = pred + (size_t)b * N * 3;
    int a0 = pairs[p * 4 + 0], a1 = pairs[p * 4 + 1];
    int a2 = pairs[p * 4 + 2], a3 = pairs[p * 4 + 3];
    float p0x = v[a0 * 3], p0y = v[a0 * 3 + 1], p0z = v[a0 * 3 + 2];
    float ex = v[a1 * 3 + 0] - p0x, ey = v[a1 * 3 + 1] - p0y, ez = v[a1 * 3 + 2] - p0z;
    float ax = v[a2 * 3 + 0] - p0x, ay = v[a2 * 3 + 1] - p0y, az = v[a2 * 3 + 2] - p0z;
    float bx = v[a3 * 3 + 0] - p0x, by = v[a3 * 3 + 1] - p0y, bz = v[a3 * 3 + 2] - p0z;
    float n0x = ey * az - ez * ay, n0y = ez * ax - ex * az, n0z = ex * ay - ey * ax;
    float n1x = -(ey * bz - ez * by), n1y = -(ez * bx - ex * bz), n1z = -(ex * by - ey * bx);
    float l0 = fmaxf(sqrtf(n0x * n0x + n0y * n0y + n0z * n0z), 1e-8f);
    float l1 = fmaxf(sqrtf(n1x * n1x + n1y * n1y + n1z * n1z), 1e-8f);
    float cosv = (n0x * n1x + n0y * n1y + n0z * n1z) / (l0 * l1);
    val = (1.0f - cosv) * scale;
  }
  wave_atomic_add(accum, val);
}

// ---- finalize -------------------------------------------------------------

__global__ void finalize_kernel(const float* __restrict__ acc,
                                float* __restrict__ out) {
  if (threadIdx.x == 0 && blockIdx.x == 0) {
    // acc[0]=cham acc[1]=edge acc[2]=lap acc[3]=nc acc[4]=vel
    out[0] = acc[0] + W_LAP * acc[2] + W_NORMAL * acc[3] +
             W_EDGE * acc[1] + W_VEL * acc[4];
  }
}

// ---- launch ---------------------------------------------------------------

extern "C" void kernel_launch(void* const* d_in, const int* in_sizes, int n_in,
                              void* d_out, int out_size, void* d_ws, size_t ws_size,
                              hipStream_t stream) {
  const float* pred = (const float*)d_in[0];
  const float* tgt  = (const float*)d_in[1];
  const int* faces  = (const int*)d_in[2];
  const int* edges  = (const int*)d_in[3];
  const int* pairs  = (const int*)d_in[4];

  const int B = BATCH;
  const int N = in_sizes[0] / (3 * B);  // 8281
  const int F = in_sizes[2] / 3;
  const int E = in_sizes[3] / 2;
  const int P = in_sizes[4] / 4;

  const int n2    = N - 1;                      // velocity point count
  const int npad  = ((N  + 15) / 16) * 16 + 16; // padded + 1 slack tile
  const int npad2 = ((n2 + 15) / 16) * 16 + 16;

  // workspace layout (floats)
  float*  accum   = (float*)d_ws;                       // [0..7]
  float*  Lx      = accum + 8;                          // B*N*3
  float*  rowsum  = Lx + (size_t)B * N * 3;             // B*N
  float4* packPP  = (float4*)(rowsum + (size_t)B * N);  // pred pos  B*npad
  float4* packTP  = packPP + (size_t)B * npad;          // tgt  pos  B*npad
  float4* packPV  = packTP + (size_t)B * npad;          // pred vel  B*npad2
  float4* packTV  = packPV + (size_t)B * npad2;         // tgt  vel  B*npad2

  // zero accumulators + laplacian scratch every launch (graph-replay safe)
  int zeroN = 8 + B * N * 3 + B * N;
  zero_kernel<<<(zeroN + 255) / 256, 256, 0, stream>>>(accum, zeroN);

  // pack points (positions and first differences) into WMMA-friendly float4s
  pack_points<<<dim3((npad  + 255) / 256, B), 256, 0, stream>>>(pred, packPP, N,  npad,  N, 0);
  pack_points<<<dim3((npad  + 255) / 256, B), 256, 0, stream>>>(tgt,  packTP, N,  npad,  N, 0);
  pack_points<<<dim3((npad2 + 255) / 256, B), 256, 0, stream>>>(pred, packPV, n2, npad2, N, 1);
  pack_points<<<dim3((npad2 + 255) / 256, B), 256, 0, stream>>>(tgt,  packTV, n2, npad2, N, 1);

  // chamfer (positions + velocities), both directions each
  {
    dim3 block(128);  // 4 waves, each owns a 16-row strip
    int rowTiles = (N + 15) / 16;
    dim3 grid((rowTiles + 3) / 4, B);
    float s = 1.0f / ((float)N * (float)B);
    chamfer_rowmin_wmma<<<grid, block, 0, stream>>>(packPP, packTP, N, npad, accum + 0, s);
    chamfer_rowmin_wmma<<<grid, block, 0, stream>>>(packTP, packPP, N, npad, accum + 0, s);

    int rowTiles2 = (n2 + 15) / 16;
    dim3 grid2((rowTiles2 + 3) / 4, B);
    float s2 = 1.0f / ((float)n2 * (float)B);
    chamfer_rowmin_wmma<<<grid2, block, 0, stream>>>(packPV, packTV, n2, npad2, accum + 4, s2);
    chamfer_rowmin_wmma<<<grid2, block, 0, stream>>>(packTV, packPV, n2, npad2, accum + 4, s2);
  }

  edge_loss_kernel<<<dim3((E + 255) / 256, B), 256, 0, stream>>>(
      pred, edges, E, N, accum + 1, 1.0f / ((float)E * (float)B));

  lap_scatter<<<dim3((F + 255) / 256, B), 256, 0, stream>>>(
      pred, faces, F, N, Lx, rowsum);
  lap_finish<<<dim3((N + 255) / 256, B), 256, 0, stream>>>(
      pred, Lx, rowsum, N, accum + 2, 1.0f / ((float)N * (float)B));

  nc_kernel<<<dim3((P + 255) / 256, B), 256, 0, stream>>>(
      pred, pairs, P, N, accum + 3, 1.0f / ((float)P * (float)B));

  finalize_kernel<<<1, 1, 0, stream>>>(accum, (float*)d_out);
}